// HashEmbedderNative_17935783428137
// MI455X (gfx1250) — compile-verified
//
#include <hip/hip_runtime.h>
#include <stdint.h>

#ifndef __has_builtin
#define __has_builtin(x) 0
#endif

// ---- gfx1250 async global<->LDS DMA paths (probe via __has_builtin) --------
#if __has_builtin(__builtin_amdgcn_global_load_async_to_lds_b32)
#define HGE_ASYNC_LOAD 1
#else
#define HGE_ASYNC_LOAD 0
#endif

#if __has_builtin(__builtin_amdgcn_global_store_async_from_lds_b128)
#define HGE_ASYNC_STORE 1
#else
#define HGE_ASYNC_STORE 0
#endif

#if HGE_ASYNC_LOAD || HGE_ASYNC_STORE
// Payload-typed pointers the builtins expect (per hipcc diagnostics):
//   b32  -> int* in AS1 / AS3
//   b128 -> int __attribute__((vector_size(16)))* in AS1 / AS3
typedef int hge_v4i __attribute__((vector_size(16)));
typedef __attribute__((address_space(1))) int* hge_g32p;
typedef __attribute__((address_space(3))) int* hge_l32p;
typedef __attribute__((address_space(1))) hge_v4i* hge_g128p;
typedef __attribute__((address_space(3))) hge_v4i* hge_l128p;

__device__ __forceinline__ hge_g32p hge_g32(const void* p) {
  return (hge_g32p)(uintptr_t)p;
}
// generic LDS pointer: low 32 bits are the LDS byte offset -> AS3 pointer
__device__ __forceinline__ hge_l32p hge_l32(const void* p) {
  return (hge_l32p)(uint32_t)(uintptr_t)p;
}
__device__ __forceinline__ hge_g128p hge_g128(const void* p) {
  return (hge_g128p)(uintptr_t)p;
}
__device__ __forceinline__ hge_l128p hge_l128(const void* p) {
  return (hge_l128p)(uint32_t)(uintptr_t)p;
}
__device__ __forceinline__ void hge_wait_async0() {
#if __has_builtin(__builtin_amdgcn_s_wait_asynccnt)
  __builtin_amdgcn_s_wait_asynccnt(0);
#else
  asm volatile("s_wait_asynccnt 0" ::: "memory");
#endif
}
#endif

#define HGE_TPB 256
#define HGE_ROWF 36  // padded row: 32 used floats + 4 pad -> 144B (16B aligned)

__global__ __launch_bounds__(HGE_TPB) void hashgrid_encode_kernel(
    const float* __restrict__ coords,
    const float* __restrict__ params,
    float* __restrict__ out,
    int npoints) {
  // level row offsets into the table (rows of 2 floats)
  constexpr uint32_t LVL_OFF[16] = {
      0u,       4096u,    36864u,   299008u,  823296u,  1347584u,
      1871872u, 2396160u, 2920448u, 3444736u, 3969024u, 4493312u,
      5017600u, 5541888u, 6066176u, 6590464u};
  constexpr uint32_t P1 = 2654435761u;  // hash prime dim1 (dim0 prime == 1)
  constexpr uint32_t P2 = 805459861u;   // hash prime dim2

  __shared__ float sout[HGE_TPB * HGE_ROWF];  // 36 KB staging for outputs

  const int t = threadIdx.x;
  const int block0 = blockIdx.x * HGE_TPB;
  const int n = block0 + t;
  const bool blk_full = (block0 + HGE_TPB) <= npoints;
  (void)blk_full;

  float cx, cy, cz;
#if HGE_ASYNC_LOAD
  __shared__ float scoords[HGE_TPB * 3];  // 3 KB
  if (blk_full) {
    // Coalesced DMA: 768 dwords, 256 lanes x 3 iterations, global -> LDS
    const float* gsrc = coords + (size_t)block0 * 3;
#pragma unroll
    for (int i = 0; i < 3; ++i) {
      const int off = t + i * HGE_TPB;
      __builtin_amdgcn_global_load_async_to_lds_b32(
          hge_g32(gsrc + off), hge_l32(&scoords[off]), 0, 0);
    }
    hge_wait_async0();   // this wave's async writes to LDS are done
    __syncthreads();     // cross-wave visibility within the block
    cx = scoords[3 * t + 0];
    cy = scoords[3 * t + 1];
    cz = scoords[3 * t + 2];
  } else
#endif
  {
    const int nc = n < npoints ? n : (npoints - 1);
    const float* cp = coords + 3 * (size_t)nc;
    cx = cp[0];
    cy = cp[1];
    cz = cp[2];
  }

  const float2* __restrict__ tbl = (const float2*)params;

#pragma unroll
  for (int l = 0; l < 16; ++l) {
    const uint32_t res = 16u << l;
    const float scale = (float)res - 1.0f;  // 16*2^l - 1, exact in fp32
    const uint32_t hs =
        (l == 0) ? 4096u : (l == 1) ? 32768u : (l == 2) ? 262144u : 524288u;
    const uint32_t mask = hs - 1u;  // all level sizes are powers of two
    const uint32_t base = LVL_OFF[l];

    const float px = cx * scale + 0.5f;
    const float py = cy * scale + 0.5f;
    const float pz = cz * scale + 0.5f;
    const float gx = floorf(px), gy = floorf(py), gz = floorf(pz);
    const float fx = px - gx, fy = py - gy, fz = pz - gz;
    const uint32_t ix = (uint32_t)(int)gx;
    const uint32_t iy = (uint32_t)(int)gy;
    const uint32_t iz = (uint32_t)(int)gz;

    // Corner bit mapping (matches reference CORNERS): bit2->x, bit1->y, bit0->z
    uint32_t r[8];
    if (l < 3) {  // dense levels: linear index x + y*res + z*res^2, then wrap
      const uint32_t res2 = res * res;
      const uint32_t b0 = ix + iy * res + iz * res2;
#pragma unroll
      for (int c = 0; c < 8; ++c) {
        const uint32_t d = ((uint32_t)((c >> 2) & 1)) +
                           ((uint32_t)((c >> 1) & 1)) * res +
                           ((uint32_t)(c & 1)) * res2;
        r[c] = ((b0 + d) & mask) + base;
      }
    } else {  // hashed levels: x ^ y*P1 ^ z*P2 (uint32 wrap == int32 wrap)
      const uint32_t hy0 = iy * P1, hy1 = hy0 + P1;
      const uint32_t hz0 = iz * P2, hz1 = hz0 + P2;
#pragma unroll
      for (int c = 0; c < 8; ++c) {
        const uint32_t hx = ix + (uint32_t)((c >> 2) & 1);
        const uint32_t hy = ((c >> 1) & 1) ? hy1 : hy0;
        const uint32_t hz = (c & 1) ? hz1 : hz0;
        r[c] = ((hx ^ hy ^ hz) & mask) + base;
      }
    }

    // Issue all 8 gathers (b64 each) before consuming: one loadcnt wait.
    float2 e[8];
#pragma unroll
    for (int c = 0; c < 8; ++c) e[c] = tbl[r[c]];

    const float wx[2] = {1.0f - fx, fx};
    const float wy[2] = {1.0f - fy, fy};
    const float wz[2] = {1.0f - fz, fz};
    float o0 = 0.0f, o1 = 0.0f;
#pragma unroll
    for (int c = 0; c < 8; ++c) {
      const float w = wx[(c >> 2) & 1] * wy[(c >> 1) & 1] * wz[c & 1];
      o0 = fmaf(w, e[c].x, o0);
      o1 = fmaf(w, e[c].y, o1);
    }
    // stage this level's 2 features in LDS (stride 36 dwords -> <=2-way conflict)
    *(float2*)&sout[t * HGE_ROWF + 2 * l] = make_float2(o0, o1);
  }

#if HGE_ASYNC_STORE
  if (blk_full) {
    __syncthreads();  // everyone's sout rows committed
    // LDS -> global DMA: per instruction a wave writes 32 consecutive 16B
    // chunks (contiguous 512B of the output tile); LDS side uses arbitrary
    // per-lane addresses into the padded rows.
    float* gdst = out + (size_t)block0 * 32;
#pragma unroll
    for (int i = 0; i < 8; ++i) {
      const int c = i * HGE_TPB + t;  // 16B chunk id within the 32KB tile
      const int p = c >> 3;           // point within block
      const int j = c & 7;            // chunk within that point's 128B row
      __builtin_amdgcn_global_store_async_from_lds_b128(
          hge_g128(gdst + (size_t)c * 4),
          hge_l128(&sout[p * HGE_ROWF + j * 4]), 0, 0);
    }
    hge_wait_async0();
  } else
#endif
  if (n < npoints) {
    float4* og = (float4*)(out + (size_t)n * 32);
#pragma unroll
    for (int i = 0; i < 8; ++i) {
      og[i] = *(float4*)&sout[t * HGE_ROWF + 4 * i];
    }
  }
}

extern "C" void kernel_launch(void* const* d_in, const int* in_sizes, int n_in,
                              void* d_out, int out_size, void* d_ws,
                              size_t ws_size, hipStream_t stream) {
  (void)n_in;
  (void)out_size;
  (void)d_ws;
  (void)ws_size;
  const float* coords = (const float*)d_in[0];  // [N,3] f32
  const float* params = (const float*)d_in[1];  // [TOTAL_ROWS*2] f32
  float* out = (float*)d_out;                   // [N,32] f32

  const int npoints = in_sizes[0] / 3;
  const int blocks = (npoints + HGE_TPB - 1) / HGE_TPB;
  hipLaunchKernelGGL(hashgrid_encode_kernel, dim3(blocks), dim3(HGE_TPB), 0,
                     stream, coords, params, out, npoints);
}